// SimpleFusionModel_67559835566302
// MI455X (gfx1250) — compile-verified
//
#include <hip/hip_runtime.h>
#include <math.h>

#define N_NODES 50000
#define FDIM    128
#define N_EDGES 600000
#define NCLS    3
#define NUM_TILES (N_NODES / 16)   // 3125, N divisible by 16 -> every tile full

typedef float v2f __attribute__((ext_vector_type(2)));
typedef float v8f __attribute__((ext_vector_type(8)));

// ---------------------------------------------------------------------------
// Kernel 1: zero the agg scratch (N*F floats) and the class-count accumulator.
// ---------------------------------------------------------------------------
__global__ __launch_bounds__(256) void zero_ws_kernel(float4* __restrict__ agg4,
                                                      float* __restrict__ counts,
                                                      int n4) {
  int i = blockIdx.x * blockDim.x + threadIdx.x;
  if (i < n4) agg4[i] = make_float4(0.f, 0.f, 0.f, 0.f);
  if (i < 4) counts[i] = 0.f;
}

// ---------------------------------------------------------------------------
// Kernel 2: segment_sum via atomics. One wave32 per edge: each lane moves a
// float4 (32 lanes * 4 = 128 features) -> coalesced 512B row read, then four
// global_atomic_add_f32 into agg[dst]. This is the roofline-dominant phase
// (~615 MB of traffic at 23.3 TB/s).
// ---------------------------------------------------------------------------
__global__ __launch_bounds__(256) void scatter_add_kernel(const float* __restrict__ x,
                                                          const int* __restrict__ ei,
                                                          float* __restrict__ agg) {
  int t = blockIdx.x * blockDim.x + threadIdx.x;
  int e = t >> 5;
  int lane = t & 31;
  if (e < N_EDGES) {
    int src = ei[e];
    int dst = ei[N_EDGES + e];
    const float4* xr = (const float4*)(x + (size_t)src * FDIM);
    float4 v = xr[lane];
    float* d = agg + (size_t)dst * FDIM + lane * 4;
    atomicAdd(d + 0, v.x);
    atomicAdd(d + 1, v.y);
    atomicAdd(d + 2, v.z);
    atomicAdd(d + 3, v.w);
  }
}

// ---------------------------------------------------------------------------
// Kernel 3: per-tile WMMA logits (fp32 16x16x4) + softmax + fusion + counts.
// One wave per 16-node tile; 8 waves (256 threads) per block.
//
// A-frag (16x4 f32): lane l -> M = l&15, K = k0 + 2*(l>>4) .. +1
// B-frag (4x16 f32): lane l -> N = l&15, same K split
// D: vgpr r -> M = r + 8*(l>>4), N = l&15
//
// W (128x3) is staged once per block into LDS padded to 16 columns of zeros,
// in a pair-major layout: element (k, n) at [ (k>>1)*32 + n*2 + (k&1) ].
// Each lane then fetches its (kb, kb+1) B pair with a single ds_load_b64,
// with no lane divergence in the hot loop.
// ---------------------------------------------------------------------------
#define WPAIR_IDX(k, n) ((((k) >> 1) * 32) + ((n) * 2) + ((k) & 1))

__global__ __launch_bounds__(256) void logits_fuse_kernel(const float* __restrict__ x,
                                                          const float* __restrict__ agg,
                                                          const float* __restrict__ Wg,
                                                          const float* __restrict__ Wa,
                                                          float* __restrict__ out,
                                                          float* __restrict__ counts) {
  __shared__ float wpad_g[FDIM * 16];  // 8 KB, pair-major padded W_gnn
  __shared__ float wpad_a[FDIM * 16];  // 8 KB, pair-major padded W_attn
  __shared__ float lg_s[8][16][4];     // [wave][node-in-tile][class]
  __shared__ float la_s[8][16][4];

  // --- stage padded weights into LDS (all 256 threads, 8 entries each) ---
  for (int idx = threadIdx.x; idx < FDIM * 16; idx += 256) {
    int kp  = idx >> 5;        // k pair index
    int rem = idx & 31;
    int n   = rem >> 1;
    int par = rem & 1;
    int k   = kp * 2 + par;
    wpad_g[idx] = (n < NCLS) ? Wg[k * NCLS + n] : 0.f;
    wpad_a[idx] = (n < NCLS) ? Wa[k * NCLS + n] : 0.f;
  }
  __syncthreads();

  const int lane = threadIdx.x & 31;
  const int wave = threadIdx.x >> 5;
  const int tile = blockIdx.x * 8 + wave;
  const int m    = lane & 15;
  const int hi   = lane >> 4;
  const bool active = (tile < NUM_TILES);   // wave-uniform

  if (active) {
    v8f cg = {};
    v8f ca = {};
    const size_t row = ((size_t)tile * 16 + m) * FDIM;
    const float* xr = x + row;
    const float* gr = agg + row;
    // this lane's LDS base for its B column: pair kp = hi at k0=0
    const float2* bg_base = (const float2*)&wpad_g[hi * 32 + m * 2];
    const float2* ba_base = (const float2*)&wpad_a[hi * 32 + m * 2];

    for (int k0 = 0; k0 < FDIM; k0 += 4) {
      const int kb = k0 + (hi << 1);
      float x0 = xr[kb], x1 = xr[kb + 1];
      float g0 = gr[kb], g1 = gr[kb + 1];
      v2f aG = {x0 + g0, x1 + g1};
      v2f aA = {x0 + 0.5f * g0, x1 + 0.5f * g1};
      // pair index advances by 2 per k0 step of 4; stride 32 floats per pair
      float2 wg2 = bg_base[(k0 >> 1) * 16 / 2 * 2];   // == bg_base[(k0>>1)*16]
      float2 wa2 = ba_base[(k0 >> 1) * 16];
      wg2 = bg_base[(k0 >> 1) * 16];
      v2f bG = {wg2.x, wg2.y};
      v2f bA = {wa2.x, wa2.y};
      // 8 args: (neg_a, A, neg_b, B, c_mod, C, reuse_a, reuse_b)
      cg = __builtin_amdgcn_wmma_f32_16x16x4_f32(false, aG, false, bG, (short)0, cg, false, false);
      ca = __builtin_amdgcn_wmma_f32_16x16x4_f32(false, aA, false, bA, (short)0, ca, false, false);
    }

    // Spill the three real logit columns to LDS (lanes with N<3 hold them).
    if (m < NCLS) {
#pragma unroll
      for (int r = 0; r < 8; ++r) {
        lg_s[wave][r + 8 * hi][m] = cg[r];
        la_s[wave][r + 8 * hi][m] = ca[r];
      }
    }
  }
  __syncthreads();

  if (active && lane < 16) {
    float gl[NCLS], al[NCLS];
#pragma unroll
    for (int c = 0; c < NCLS; ++c) {
      gl[c] = lg_s[wave][lane][c];
      al[c] = la_s[wave][lane][c];
    }
    // stable softmax over 3 classes, both heads
    float gmxl = fmaxf(gl[0], fmaxf(gl[1], gl[2]));
    float amxl = fmaxf(al[0], fmaxf(al[1], al[2]));
    float g[NCLS], a[NCLS];
    float gsum = 0.f, asum = 0.f;
#pragma unroll
    for (int c = 0; c < NCLS; ++c) {
      g[c] = expf(gl[c] - gmxl); gsum += g[c];
      a[c] = expf(al[c] - amxl); asum += a[c];
    }
    float gi = 1.f / gsum, ai = 1.f / asum;
#pragma unroll
    for (int c = 0; c < NCLS; ++c) { g[c] *= gi; a[c] *= ai; }

    // argmax/max (first-index-on-tie like jnp.argmax)
    int gpred = 0; float gmax = g[0];
    if (g[1] > gmax) { gmax = g[1]; gpred = 1; }
    if (g[2] > gmax) { gmax = g[2]; gpred = 2; }
    int apred = 0; float amax = a[0];
    if (a[1] > amax) { amax = a[1]; apred = 1; }
    if (a[2] > amax) { amax = a[2]; apred = 2; }

    const float cw0[NCLS]   = {0.70f, 0.88f, 0.18f};
    const float cw1[NCLS]   = {0.30f, 0.12f, 0.82f};
    const float boost[NCLS] = {1.05f, 1.04f, 1.02f};

    bool  gs  = gmax > amax;
    float tot = gmax + amax;
    float r   = fminf(fmaxf((gs ? gmax : amax) / tot, 0.6f), 0.85f);
    bool consensus = (gpred == apred);
    bool high_conf = (gmax > 0.85f) || (amax > 0.85f);
    int  hc_class  = gs ? gpred : apred;

    float comb[NCLS];
    float s = 0.f;
#pragma unroll
    for (int c = 0; c < NCLS; ++c) {
      float base = g[c] * cw0[c] + a[c] * cw1[c];
      float cons = (gs ? (r * g[c] + (1.f - r) * a[c])
                       : ((1.f - r) * g[c] + r * a[c])) * 1.1f;
      float hcv  = gs ? (0.95f * g[c] + 0.05f * a[c])
                      : (0.05f * g[c] + 0.95f * a[c]);
      float v = consensus ? ((c == gpred) ? cons : base)
                          : (high_conf ? ((c == hc_class) ? hcv : base)
                                       : base * boost[c]);
      comb[c] = v;
      s += v;
    }
    float inv = 1.f / s;
    size_t o = ((size_t)tile * 16 + lane) * NCLS;
    int pred = 0; float pm = comb[0] * inv;
#pragma unroll
    for (int c = 0; c < NCLS; ++c) {
      comb[c] *= inv;
      out[o + c] = comb[c];
    }
    if (comb[1] > pm) { pm = comb[1]; pred = 1; }
    if (comb[2] > pm) { pm = comb[2]; pred = 2; }
    atomicAdd(&counts[pred], 1.0f);
  }
}

// ---------------------------------------------------------------------------
// Kernel 4: compute per-class adjust factors from the global histogram.
// ---------------------------------------------------------------------------
__global__ void adjust_kernel(const float* __restrict__ counts,
                              float* __restrict__ adjust) {
  if (threadIdx.x == 0 && blockIdx.x == 0) {
    const float ideal[NCLS] = {0.33f, 0.33f, 0.34f};
#pragma unroll
    for (int c = 0; c < NCLS; ++c) {
      float v = ideal[c] / (counts[c] / (float)N_NODES + 1e-6f);
      adjust[c] = fminf(fmaxf(v, 0.95f), 1.05f);
    }
  }
}

// ---------------------------------------------------------------------------
// Kernel 5: apply adjust and renormalize rows (in place on d_out).
// ---------------------------------------------------------------------------
__global__ __launch_bounds__(256) void finalize_kernel(float* __restrict__ out,
                                                       const float* __restrict__ adjust) {
  int i = blockIdx.x * blockDim.x + threadIdx.x;
  if (i < N_NODES) {
    float a0 = adjust[0], a1 = adjust[1], a2 = adjust[2];
    size_t o = (size_t)i * NCLS;
    float v0 = out[o + 0] * a0;
    float v1 = out[o + 1] * a1;
    float v2 = out[o + 2] * a2;
    float inv = 1.f / (v0 + v1 + v2);
    out[o + 0] = v0 * inv;
    out[o + 1] = v1 * inv;
    out[o + 2] = v2 * inv;
  }
}

// ---------------------------------------------------------------------------
extern "C" void kernel_launch(void* const* d_in, const int* in_sizes, int n_in,
                              void* d_out, int out_size, void* d_ws, size_t ws_size,
                              hipStream_t stream) {
  const float* x  = (const float*)d_in[0];   // (N, F)
  const int*   ei = (const int*)d_in[1];     // (2, E) row-major: src then dst
  const float* Wg = (const float*)d_in[2];   // (F, C)
  const float* Wa = (const float*)d_in[3];   // (F, C)
  float* out = (float*)d_out;                // (N, C)

  char* ws = (char*)d_ws;
  float* agg    = (float*)ws;                                            // N*F floats
  float* counts = (float*)(ws + (size_t)N_NODES * FDIM * sizeof(float)); // 4 floats
  float* adjust = counts + 4;                                            // 4 floats

  // 1) zero agg + counts
  int n4 = N_NODES * FDIM / 4;  // 1,600,000 float4s
  zero_ws_kernel<<<(n4 + 255) / 256, 256, 0, stream>>>((float4*)agg, counts, n4);

  // 2) edge scatter-add (one wave32 per edge)
  scatter_add_kernel<<<(N_EDGES * 32) / 256, 256, 0, stream>>>(x, ei, agg);

  // 3) WMMA logits + softmax + fusion + class histogram
  int nblk = (NUM_TILES + 7) / 8;  // 391 blocks of 8 waves
  logits_fuse_kernel<<<nblk, 256, 0, stream>>>(x, agg, Wg, Wa, out, counts);

  // 4) class-balance adjustment factors
  adjust_kernel<<<1, 32, 0, stream>>>(counts, adjust);

  // 5) apply + renormalize
  finalize_kernel<<<(N_NODES + 255) / 256, 256, 0, stream>>>(out, adjust);
}